// ChamferLoss_44856638440163
// MI455X (gfx1250) — compile-verified
//
#include <hip/hip_runtime.h>
#include <math.h>

typedef __attribute__((ext_vector_type(2))) float v2f;
typedef __attribute__((ext_vector_type(8))) float v8f;

// ---------------------------------------------------------------------------
// Kernel 1: unproject depth image through camera -> packed pred [px,py,pz,1]
// and squared norms x2[n].
// ---------------------------------------------------------------------------
__global__ void __launch_bounds__(256)
rays_kernel(const float* __restrict__ c, const float* __restrict__ depth,
            float* __restrict__ predA, float* __restrict__ x2,
            int B, int N, int res) {
    int idx = blockIdx.x * blockDim.x + threadIdx.x;
    if (idx >= B * N) return;
    int b = idx / N;
    int n = idx - b * N;

    const float* cb = c + b * 25;
    // intrinsics (row-major 3x3 at offset 16)
    float fx = cb[16], sk = cb[17], cx = cb[18];
    float fy = cb[20], cy = cb[21];

    int i = n / res;
    int j = n - i * res;
    float inv = 1.0f / (float)res;
    float x_cam = ((float)j + 0.5f) * inv;
    float y_cam = ((float)i + 0.5f) * inv;

    float x_lift = (x_cam - cx + cy * sk / fy - sk * y_cam / fy) / fx;
    float y_lift = (y_cam - cy) / fy;

    // dir = world - origin = R * [x_lift, y_lift, 1]   (cam2world row-major 4x4)
    float dx = cb[0] * x_lift + cb[1] * y_lift + cb[2];
    float dy = cb[4] * x_lift + cb[5] * y_lift + cb[6];
    float dz = cb[8] * x_lift + cb[9] * y_lift + cb[10];
    float ilen = 1.0f / sqrtf(dx * dx + dy * dy + dz * dz);

    float d = depth[idx];
    float px = d * dx * ilen + cb[3];
    float py = d * dy * ilen + cb[7];
    float pz = d * dz * ilen + cb[11];

    float4 o; o.x = px; o.y = py; o.z = pz; o.w = 1.0f;
    ((float4*)predA)[idx] = o;
    x2[idx] = px * px + py * py + pz * pz;
}

// ---------------------------------------------------------------------------
// Kernel 2: pack point cloud as B-matrix operand [-2x,-2y,-2z, |q|^2]
// ---------------------------------------------------------------------------
__global__ void __launch_bounds__(256)
pcprep_kernel(const float* __restrict__ pc, float* __restrict__ pcB, int total) {
    int idx = blockIdx.x * blockDim.x + threadIdx.x;
    if (idx >= total) return;
    float x = pc[idx * 3 + 0];
    float y = pc[idx * 3 + 1];
    float z = pc[idx * 3 + 2];
    float4 o;
    o.x = -2.0f * x; o.y = -2.0f * y; o.z = -2.0f * z;
    o.w = x * x + y * y + z * z;
    ((float4*)pcB)[idx] = o;
}

// ---------------------------------------------------------------------------
// Kernel 3: chamfer min via V_WMMA_F32_16X16X4_F32.
// One wave32 per 16-row n-tile. A = [p,1] (16x4), B = [-2q,|q|^2] (4x16).
// WMMA tile c[n][m] = |q_m|^2 - 2<p_n,q_m>; running elementwise min over all
// m-tiles, then xor-shuffle min across the 16 lanes of each half, add |p_n|^2.
//
// A-operand layout (16x4 f32, M=lane%16): vgpr g, lane half h -> K = 2*h+g.
// B-operand layout (4x16 f32, N=lane%16): vgpr g, lane half h -> K = 2*h+g.
// C/D layout (16x16 f32): vgpr r, lanes 0-15 -> M=r, lanes 16-31 -> M=r+8.
// ---------------------------------------------------------------------------
__global__ void __launch_bounds__(256)
chamfer_min_kernel(const float* __restrict__ predA, const float* __restrict__ x2,
                   const float* __restrict__ pcB, float* __restrict__ d0,
                   int N, int M) {
    int wave  = (blockIdx.x * blockDim.x + threadIdx.x) >> 5;
    int lane  = threadIdx.x & 31;
    int nTiles = N >> 4;
    int b     = wave / nTiles;
    int ntile = wave - b * nTiles;
    int half  = lane >> 4;   // 0: K=0,1   1: K=2,3
    int l16   = lane & 15;

    // Load A operand once (stays in VGPRs for all 256 WMMAs).
    const float* ap = predA + ((size_t)(b * N + ntile * 16 + l16) * 4 + half * 2);
    v2f a; a.x = ap[0]; a.y = ap[1];

    v8f vmin;
#pragma unroll
    for (int r = 0; r < 8; ++r) vmin[r] = 3.0e38f;

    const float* pb = pcB + (size_t)b * M * 4;
    int mTiles = M >> 4;
#pragma unroll 4
    for (int mt = 0; mt < mTiles; ++mt) {
        const float* bp = pb + ((size_t)(mt * 16 + l16) * 4 + half * 2);
        v2f bv; bv.x = bp[0]; bv.y = bp[1];

        v8f cz = {};
        // 8 args: (neg_a, A, neg_b, B, c_mod, C, reuse_a, reuse_b)
        v8f cacc = __builtin_amdgcn_wmma_f32_16x16x4_f32(
            false, a, false, bv, (short)0, cz, false, false);
#pragma unroll
        for (int r = 0; r < 8; ++r) vmin[r] = fminf(vmin[r], cacc[r]);
    }

    // Min across the 16 lanes of each half (columns m of the tile).
#pragma unroll
    for (int off = 1; off < 16; off <<= 1) {
#pragma unroll
        for (int r = 0; r < 8; ++r) {
            float o = __shfl_xor(vmin[r], off, 16);
            vmin[r] = fminf(vmin[r], o);
        }
    }

    // lanes 0 / 16 hold mins for rows ntile*16 + r (+8 for upper half).
    if (l16 == 0) {
        int base = b * N + ntile * 16 + half * 8;
#pragma unroll
        for (int r = 0; r < 8; ++r)
            d0[base + r] = vmin[r] + x2[base + r];
    }
}

// ---------------------------------------------------------------------------
// Kernel 4: per-batch bitonic sort of d0 (N=4096 in LDS), mean of smallest K,
// output 2*mean (reference adds d0-derived loss twice).
// ---------------------------------------------------------------------------
__global__ void __launch_bounds__(1024)
topk_mean_kernel(const float* __restrict__ d0, float* __restrict__ out,
                 int N, int K) {
    __shared__ float s[4096];
    __shared__ float red[1024];
    int b = blockIdx.x;
    int tid = threadIdx.x;

    for (int i = tid; i < N; i += 1024) s[i] = d0[b * N + i];
    __syncthreads();

    for (int k = 2; k <= N; k <<= 1) {
        for (int j = k >> 1; j > 0; j >>= 1) {
            for (int i = tid; i < N; i += 1024) {
                int ixj = i ^ j;
                if (ixj > i) {
                    bool up = ((i & k) == 0);
                    float va = s[i], vb = s[ixj];
                    if ((va > vb) == up) { s[i] = vb; s[ixj] = va; }
                }
            }
            __syncthreads();
        }
    }

    float acc = 0.0f;
    for (int i = tid; i < K; i += 1024) acc += s[i];
    red[tid] = acc;
    __syncthreads();
    for (int step = 512; step > 0; step >>= 1) {
        if (tid < step) red[tid] += red[tid + step];
        __syncthreads();
    }
    if (tid == 0) out[b] = 2.0f * red[0] / (float)K;
}

// ---------------------------------------------------------------------------
extern "C" void kernel_launch(void* const* d_in, const int* in_sizes, int n_in,
                              void* d_out, int out_size, void* d_ws, size_t ws_size,
                              hipStream_t stream) {
    const float* c     = (const float*)d_in[0];   // (B,25)
    const float* depth = (const float*)d_in[1];   // (B,N)
    const float* pc    = (const float*)d_in[2];   // (B,M,3)

    int B = in_sizes[0] / 25;
    int N = in_sizes[1] / B;
    int M = in_sizes[2] / (3 * B);
    int res = 1;
    while (res * res < N) ++res;                  // 64
    int K = ((N < M ? N : M) / 2);                // 2048

    // Workspace layout (floats). B*N*5 is 16B-aligned for B=8,N=4096.
    float* ws    = (float*)d_ws;
    float* predA = ws;                              // B*N*4
    float* x2    = predA + (size_t)B * N * 4;       // B*N
    float* pcB   = x2    + (size_t)B * N;           // B*M*4
    float* dmin  = pcB   + (size_t)B * M * 4;       // B*N

    int t1 = B * N;
    rays_kernel<<<(t1 + 255) / 256, 256, 0, stream>>>(c, depth, predA, x2, B, N, res);

    int t2 = B * M;
    pcprep_kernel<<<(t2 + 255) / 256, 256, 0, stream>>>(pc, pcB, t2);

    int waves = B * (N / 16);                       // 2048 waves
    int blocks = (waves * 32) / 256;                // 8 waves/block
    chamfer_min_kernel<<<blocks, 256, 0, stream>>>(predA, x2, pcB, dmin, N, M);

    topk_mean_kernel<<<B, 1024, 0, stream>>>(dmin, (float*)d_out, N, K);
}